// ProductionPaperQNN_33681133535932
// MI455X (gfx1250) — compile-verified
//
#include <hip/hip_runtime.h>
#include <hip/hip_bf16.h>
#include <math.h>

// ---------------------------------------------------------------------------
// QNN batch collapse:
//   q[b] = psi_b^T * M * psi_b,  M = Re(U^H D U) fixed 256x256 (shared params)
//   psi_b = real rank-1 tensor product from 8 FRQI angles (4 quadrant means)
// Pipeline: build_u -> build_m -> build_psi -> quad_form (WMMA f32) -> mlp
// ---------------------------------------------------------------------------

typedef float v2f __attribute__((ext_vector_type(2)));
typedef float v8f __attribute__((ext_vector_type(8)));

// ----------------------------- circuit helpers -----------------------------
// State: 256 complex amplitudes in LDS, 128 threads. Qubit w <-> bit (7-w).

__device__ __forceinline__ void gate_ry(float2* st, int w, float theta, int tid) {
    int mask = 1 << (7 - w);
    float s, c;
    sincosf(theta * 0.5f, &s, &c);
    int k0 = ((tid & ~(mask - 1)) << 1) | (tid & (mask - 1));
    int k1 = k0 | mask;
    float2 a = st[k0], b = st[k1];
    st[k0] = make_float2(c * a.x - s * b.x, c * a.y - s * b.y);
    st[k1] = make_float2(s * a.x + c * b.x, s * a.y + c * b.y);
    __syncthreads();
}

__device__ __forceinline__ void gate_rz(float2* st, int w, float theta, int tid) {
    int mask = 1 << (7 - w);
    float sh, ch;
    sincosf(theta * 0.5f, &sh, &ch);
    for (int k = tid; k < 256; k += 128) {
        float2 a = st[k];
        float ps = (k & mask) ? sh : -sh;   // exp(+i h) for bit=1, exp(-i h) for bit=0
        st[k] = make_float2(a.x * ch - a.y * ps, a.x * ps + a.y * ch);
    }
    __syncthreads();
}

__device__ __forceinline__ void gate_cnot(float2* st, int c, int t, int tid) {
    int mc = 1 << (7 - c), mt = 1 << (7 - t);
    for (int k = tid; k < 256; k += 128) {
        if ((k & mc) && !(k & mt)) {
            float2 a = st[k], b = st[k | mt];
            st[k] = b;
            st[k | mt] = a;
        }
    }
    __syncthreads();
}

// Column j of U: apply the shared parameterized circuit to basis state e_j.
__global__ void __launch_bounds__(128) build_u(const float* __restrict__ cp,
                                               float* __restrict__ Ur,
                                               float* __restrict__ Ui) {
    __shared__ float2 st[256];
    const int tid = threadIdx.x;
    const int j = blockIdx.x;
    for (int k = tid; k < 256; k += 128)
        st[k] = make_float2((k == j) ? 1.f : 0.f, 0.f);
    __syncthreads();

    int idx = 0;
    for (int layer = 0; layer < 2; ++layer) {
        for (int i = 0; i < 7; ++i) {
            gate_ry(st, i,     cp[idx + 0],  tid);
            gate_rz(st, i,     cp[idx + 1],  tid);
            gate_ry(st, i + 1, cp[idx + 2],  tid);
            gate_rz(st, i + 1, cp[idx + 3],  tid);
            gate_cnot(st, i, i + 1, tid);
            gate_ry(st, i,     cp[idx + 4],  tid);
            gate_rz(st, i,     cp[idx + 5],  tid);
            gate_ry(st, i + 1, cp[idx + 6],  tid);
            gate_rz(st, i + 1, cp[idx + 7],  tid);
            gate_cnot(st, i + 1, i, tid);
            gate_ry(st, i,     cp[idx + 8],  tid);
            gate_rz(st, i,     cp[idx + 9],  tid);
            gate_ry(st, i + 1, cp[idx + 10], tid);
            gate_rz(st, i + 1, cp[idx + 11], tid);
            gate_cnot(st, i, i + 1, tid);
            gate_ry(st, i,     cp[idx + 12], tid);
            gate_ry(st, i + 1, cp[idx + 13], tid);
            idx += 14;
        }
    }
    gate_ry(st, 0, cp[idx], tid);

    for (int k = tid; k < 256; k += 128) {
        Ur[k * 256 + j] = st[k].x;
        Ui[k * 256 + j] = st[k].y;
    }
}

// M[i][j] = sum_k sign(k) * (Ur[k][i]Ur[k][j] + Ui[k][i]Ui[k][j]), sign = +1 for k<128.
__global__ void __launch_bounds__(256) build_m(const float* __restrict__ Ur,
                                               const float* __restrict__ Ui,
                                               float* __restrict__ M) {
    const int i = blockIdx.x;
    const int j = threadIdx.x;
    float acc = 0.f;
    for (int k = 0; k < 256; ++k) {
        float sgn = (k < 128) ? 1.f : -1.f;
        acc += sgn * (Ur[k * 256 + i] * Ur[k * 256 + j] +
                      Ui[k * 256 + i] * Ui[k * 256 + j]);
    }
    M[i * 256 + j] = acc;
}

// FRQI encode + tensor-product state. One block per image, 256 threads.
__global__ void __launch_bounds__(256) build_psi(const float* __restrict__ x,
                                                 float* __restrict__ Psi) {
    __shared__ float red[4 * 256];
    __shared__ float cs[16];        // (c_i, s_i) for i = 0..7
    const int tid = threadIdx.x;
    const int b = blockIdx.x;
    const float* img = x + (size_t)b * 784;

    float part[4] = {0.f, 0.f, 0.f, 0.f};
    for (int p = tid; p < 784; p += 256) {
        int r = p / 28, c = p - r * 28;
        int quad = ((r >= 14) ? 2 : 0) + ((c >= 14) ? 1 : 0);
        part[quad] += img[p];
    }
    for (int qd = 0; qd < 4; ++qd) red[qd * 256 + tid] = part[qd];
    __syncthreads();
    for (int s = 128; s > 0; s >>= 1) {           // deterministic tree reduce
        if (tid < s)
            for (int qd = 0; qd < 4; ++qd)
                red[qd * 256 + tid] += red[qd * 256 + tid + s];
        __syncthreads();
    }
    if (tid < 8) {
        const float m0 = red[0]       * (1.f / 196.f);   // TL quadrant mean
        const float m1 = red[256]     * (1.f / 196.f);   // TR
        const float m2 = red[2 * 256] * (1.f / 196.f);   // BL
        const float m3 = red[3 * 256] * (1.f / 196.f);   // BR
        float reg;
        switch (tid) {
            case 0: reg = m0; break;
            case 1: reg = m1; break;
            case 2: reg = m2; break;
            case 3: reg = m3; break;
            case 4: reg = 0.5f * (m0 + m2); break;   // left half
            case 5: reg = 0.5f * (m1 + m3); break;   // right half
            case 6: reg = 0.5f * (m0 + m1); break;   // top half
            default: reg = 0.5f * (m2 + m3); break;  // bottom half
        }
        const float PI_F = 3.14159265358979323846f;
        float th = fminf(fmaxf((reg + 1.f) * (PI_F * 0.25f), 0.f), PI_F * 0.5f);
        float sv, cv;
        sincosf(th * 0.5f, &sv, &cv);
        cs[2 * tid]     = cv;
        cs[2 * tid + 1] = sv;
    }
    __syncthreads();

    // psi[k] = prod_i (bit_i(k) ? s_i : c_i), bit_i(k) = (k >> (7-i)) & 1
    float v = 1.f;
    #pragma unroll
    for (int i = 0; i < 8; ++i) {
        int bit = (tid >> (7 - i)) & 1;
        v *= cs[2 * i + bit];
    }
    Psi[(size_t)b * 256 + tid] = v;
}

// q[b] = psi_b^T M psi_b via V_WMMA_F32_16X16X4_F32.
// 16 batch rows per workgroup; 4 waves each own 4 of the 16 j-tiles.
#define PSI_STRIDE 260   // 256 + 4 pad: conflict-free LDS (bank = 4*lane + k mod 64)

__global__ void __launch_bounds__(128) quad_form(const float* __restrict__ Psi,
                                                 const float* __restrict__ M,
                                                 float* __restrict__ q) {
    __shared__ float psiL[16 * PSI_STRIDE];
    __shared__ float Tls [16 * PSI_STRIDE];
    __shared__ float red2[16 * 8];
    const int tid = threadIdx.x;
    const int b0 = blockIdx.x * 16;

    for (int e = tid; e < 16 * 256; e += 128) {
        int r = e >> 8, c = e & 255;
        psiL[r * PSI_STRIDE + c] = Psi[(size_t)(b0 + r) * 256 + c];
    }
    __syncthreads();

    const int lane = tid & 31;
    const int wave = tid >> 5;
    const int half = lane >> 4;      // 0: K0/K1 slice, 1: K2/K3 slice
    const int l16  = lane & 15;

    for (int tt = 0; tt < 4; ++tt) {
        const int jbase = (wave * 4 + tt) * 16;
        v8f acc = {};
        #pragma unroll 4
        for (int kk = 0; kk < 64; ++kk) {
            const int k0 = kk * 4 + half * 2;
            v2f a, bm;
            // A (16x4): lanes 0-15 rows, VGPR0 = K{0,2}, VGPR1 = K{1,3}
            a.x = psiL[l16 * PSI_STRIDE + k0];
            a.y = psiL[l16 * PSI_STRIDE + k0 + 1];
            // B (4x16): row k striped across lanes; VGPR0 = K{0,2}, VGPR1 = K{1,3}
            bm.x = M[(size_t)k0 * 256 + jbase + l16];
            bm.y = M[(size_t)(k0 + 1) * 256 + jbase + l16];
            acc = __builtin_amdgcn_wmma_f32_16x16x4_f32(
                false, a, false, bm, (short)0, acc, false, false);
        }
        // C/D 16x16: VGPR p, lanes 0-15 -> row p, lanes 16-31 -> row p+8
        #pragma unroll
        for (int p = 0; p < 8; ++p) {
            int row = p + half * 8;
            Tls[row * PSI_STRIDE + jbase + l16] = acc[p];
        }
    }
    __syncthreads();

    // q[row] = sum_c T[row][c] * psi[row][c], deterministic 2-stage reduce
    {
        const int row = tid >> 3;    // 0..15
        const int sub = tid & 7;     // 0..7
        float acc = 0.f;
        for (int c = sub * 32; c < sub * 32 + 32; ++c)
            acc += Tls[row * PSI_STRIDE + c] * psiL[row * PSI_STRIDE + c];
        red2[row * 8 + sub] = acc;
    }
    __syncthreads();
    if (tid < 16) {
        float acc = 0.f;
        for (int s = 0; s < 8; ++s) acc += red2[tid * 8 + s];
        q[b0 + tid] = acc;
    }
}

// Tiny MLP head: 1 -> 32 (relu, affine) -> 16 (relu) -> 1. One thread per image.
__global__ void __launch_bounds__(256) mlp_head(const float* __restrict__ q,
                                                const float* __restrict__ W1,
                                                const float* __restrict__ b1,
                                                const float* __restrict__ gamma,
                                                const float* __restrict__ beta,
                                                const float* __restrict__ W2,
                                                const float* __restrict__ b2,
                                                const float* __restrict__ W3,
                                                const float* __restrict__ b3,
                                                float* __restrict__ out, int B) {
    const int b = blockIdx.x * blockDim.x + threadIdx.x;
    if (b >= B) return;
    const float qv = q[b];
    const float inv = rsqrtf(1.f + 1e-5f);
    float h1[32];
    #pragma unroll
    for (int j = 0; j < 32; ++j) {
        float h = fmaxf(qv * W1[j] + b1[j], 0.f);
        h1[j] = h * (gamma[j] * inv) + beta[j];
    }
    float o = b3[0];
    #pragma unroll
    for (int j2 = 0; j2 < 16; ++j2) {
        float acc = b2[j2];
        #pragma unroll
        for (int j = 0; j < 32; ++j) acc += h1[j] * W2[j2 * 32 + j];
        o += fmaxf(acc, 0.f) * W3[j2];
    }
    out[b] = o;
}

// ---------------------------------------------------------------------------

extern "C" void kernel_launch(void* const* d_in, const int* in_sizes, int n_in,
                              void* d_out, int out_size, void* d_ws, size_t ws_size,
                              hipStream_t stream) {
    const float* x     = (const float*)d_in[0];
    const float* cp    = (const float*)d_in[1];
    const float* W1    = (const float*)d_in[2];
    const float* b1    = (const float*)d_in[3];
    const float* gamma = (const float*)d_in[4];
    const float* beta  = (const float*)d_in[5];
    const float* W2    = (const float*)d_in[6];
    const float* b2    = (const float*)d_in[7];
    const float* W3    = (const float*)d_in[8];
    const float* b3    = (const float*)d_in[9];
    float* out = (float*)d_out;

    const int B = in_sizes[0] / 784;     // 4096

    float* ws  = (float*)d_ws;
    float* Ur  = ws;                       // 256*256
    float* Ui  = Ur  + 256 * 256;          // 256*256
    float* Mm  = Ui  + 256 * 256;          // 256*256
    float* Psi = Mm  + 256 * 256;          // B*256
    float* q   = Psi + (size_t)B * 256;    // B

    build_u  <<<256, 128, 0, stream>>>(cp, Ur, Ui);
    build_m  <<<256, 256, 0, stream>>>(Ur, Ui, Mm);
    build_psi<<<B,   256, 0, stream>>>(x, Psi);
    quad_form<<<B / 16, 128, 0, stream>>>(Psi, Mm, q);
    mlp_head <<<(B + 255) / 256, 256, 0, stream>>>(q, W1, b1, gamma, beta,
                                                   W2, b2, W3, b3, out, B);
}